// AliasFreeActivation_86517821211774
// MI455X (gfx1250) — compile-verified
//
#include <hip/hip_runtime.h>

// Alias-free activation (upfirdn 2x up -> lrelu -> 2x down), fused, fp32.
// N=8, C=128, H=W=128 ; up/down filters 12x12 ; out 127x127 per plane.

#define TILE        32      // output tile edge per workgroup
#define IN_T        43      // staged input tile edge  (TILE+11)
#define IN_STRIDE   44
#define MPH         37      // midphase tile edge per phase (TILE+5)
#define MPH_STRIDE  38

typedef __attribute__((address_space(1))) int g_int;   // global (device) int
typedef __attribute__((address_space(3))) int l_int;   // LDS int

// ---- CDNA5 async global->LDS DMA (ASYNCcnt path) -------------------------
__device__ __forceinline__ void async_g2l_b32(const float* g, float* l) {
#if __has_builtin(__builtin_amdgcn_global_load_async_to_lds_b32)
  __builtin_amdgcn_global_load_async_to_lds_b32(
      (g_int*)g, (l_int*)l, /*imm offset*/0, /*cpol*/0);
#else
  unsigned lds_addr = (unsigned)(unsigned long long)(l_int*)l;
  unsigned long long ga = (unsigned long long)g;
  asm volatile("global_load_async_to_lds_b32 %0, %1, off"
               :: "v"(lds_addr), "v"(ga) : "memory");
#endif
}

__device__ __forceinline__ void wait_async_all() {
#if __has_builtin(__builtin_amdgcn_s_wait_asynccnt)
  __builtin_amdgcn_s_wait_asynccnt(0);
#else
  asm volatile("s_wait_asynccnt 0x0" ::: "memory");
#endif
}

__global__ __launch_bounds__(256)
void aliasfree_upfirdn_fused(const float* __restrict__ x,
                             const float* __restrict__ bias,
                             const float* __restrict__ fu,
                             const float* __restrict__ fd,
                             float* __restrict__ out)
{
  __shared__ float in_t[IN_T * IN_STRIDE + 16];          // staged input tile (+bias)
  __shared__ float mph[4 * MPH * MPH_STRIDE + 16];       // 4 polyphase mid planes
  __shared__ float fu_l[144];
  __shared__ float fd_l[144];

  const int tid   = threadIdx.x;
  const int plane = blockIdx.z;          // n*128 + c
  const int c     = plane & 127;
  const int x0    = blockIdx.x * TILE;
  const int y0    = blockIdx.y * TILE;

  // ---- stage 0a: async-DMA both 12x12 filters into LDS (288 words) --------
  for (int i = tid; i < 288; i += 256) {
    if (i < 144) async_g2l_b32(fu + i,         &fu_l[i]);
    else         async_g2l_b32(fd + (i - 144), &fd_l[i - 144]);
  }

  const float b = bias[c];
  const float* __restrict__ xin = x + (size_t)plane * (128 * 128);

  // ---- stage 0b: async-DMA input halo tile [y0-5, y0+37] x [x0-5, x0+37] ---
  for (int idx = tid; idx < IN_T * IN_T; idx += 256) {
    const int rr = idx / IN_T;
    const int cc = idx - rr * IN_T;
    const int gy = y0 - 5 + rr;
    const int gx = x0 - 5 + cc;
    float* dst = &in_t[rr * IN_STRIDE + cc];
    if ((unsigned)gy < 128u && (unsigned)gx < 128u) {
      async_g2l_b32(xin + (gy << 7) + gx, dst);
    } else {
      *dst = 0.0f;                       // zero padding of the conv
    }
  }
  wait_async_all();
  __syncthreads();

  // fuse per-channel bias into real samples only (padding must stay zero)
  for (int idx = tid; idx < IN_T * IN_T; idx += 256) {
    const int rr = idx / IN_T;
    const int cc = idx - rr * IN_T;
    const int gy = y0 - 5 + rr;
    const int gx = x0 - 5 + cc;
    if ((unsigned)gy < 128u && (unsigned)gx < 128u)
      in_t[rr * IN_STRIDE + cc] += b;
  }
  __syncthreads();

  // ---- stage 1: polyphase up-FIR + gain + lrelu + clamp -> mph -------------
  // mph[r,s][a,b] = act( 4 * sum_{i',j'<6} fu[2i'+r][2j'+s] * in_t[a+i'+r][b+j'+s] )
  // 4 phases * 37 rows * 10 strips-of-4 = 1480 strips
  for (int sidx = tid; sidx < 4 * MPH * 10; sidx += 256) {
    const int ph  = sidx / (MPH * 10);
    const int rem = sidx - ph * (MPH * 10);
    const int la  = rem / 10;
    const int lb0 = (rem - la * 10) << 2;
    const int r = ph >> 1, s = ph & 1;

    float a0 = 0.f, a1 = 0.f, a2 = 0.f, a3 = 0.f;
#pragma unroll
    for (int ip = 0; ip < 6; ++ip) {
      const float* row = &in_t[(la + ip + r) * IN_STRIDE + lb0 + s];
      const float* fr  = &fu_l[(2 * ip + r) * 12 + s];
#pragma unroll
      for (int jp = 0; jp < 6; ++jp) {
        const float f = fr[2 * jp];
        a0 = fmaf(f, row[jp + 0], a0);
        a1 = fmaf(f, row[jp + 1], a1);
        a2 = fmaf(f, row[jp + 2], a2);
        a3 = fmaf(f, row[jp + 3], a3);
      }
    }
    float vals[4] = {a0, a1, a2, a3};
#pragma unroll
    for (int k = 0; k < 4; ++k) {
      if (lb0 + k < MPH) {
        float v = 4.0f * vals[k];                        // UP*UP gain
        v = (v >= 0.0f ? v : 0.2f * v) * 1.41421356237f; // lrelu * sqrt(2)
        v = fminf(fmaxf(v, -256.0f), 256.0f);            // clamp
        mph[(ph * MPH + la) * MPH_STRIDE + lb0 + k] = v;
      }
    }
  }
  __syncthreads();

  // ---- stage 2: polyphase down-FIR, one 1x4 output strip per thread --------
  const int ly  = tid >> 3;         // 0..31
  const int lx0 = (tid & 7) << 2;   // 0,4,...,28
  float o0 = 0.f, o1 = 0.f, o2 = 0.f, o3 = 0.f;
#pragma unroll
  for (int ph = 0; ph < 4; ++ph) {
    const int q = ph >> 1, s = ph & 1;
#pragma unroll
    for (int ip = 0; ip < 6; ++ip) {
      const float* row = &mph[(ph * MPH + ly + ip) * MPH_STRIDE + lx0];
      const float* fr  = &fd_l[(2 * ip + q) * 12 + s];
#pragma unroll
      for (int jp = 0; jp < 6; ++jp) {
        const float f = fr[2 * jp];
        o0 = fmaf(f, row[jp + 0], o0);
        o1 = fmaf(f, row[jp + 1], o1);
        o2 = fmaf(f, row[jp + 2], o2);
        o3 = fmaf(f, row[jp + 3], o3);
      }
    }
  }

  const int gy = y0 + ly;
  if (gy < 127) {
    const size_t base = (size_t)plane * (127 * 127) + (size_t)gy * 127;
    const float ov[4] = {o0, o1, o2, o3};
#pragma unroll
    for (int k = 0; k < 4; ++k) {
      const int gx = x0 + lx0 + k;
      if (gx < 127) out[base + gx] = ov[k];
    }
  }
}

extern "C" void kernel_launch(void* const* d_in, const int* in_sizes, int n_in,
                              void* d_out, int out_size, void* d_ws, size_t ws_size,
                              hipStream_t stream) {
  const float* x    = (const float*)d_in[0];
  const float* bias = (const float*)d_in[1];
  const float* fu   = (const float*)d_in[2];
  const float* fd   = (const float*)d_in[3];
  float* out        = (float*)d_out;

  dim3 grid((127 + TILE - 1) / TILE, (127 + TILE - 1) / TILE, 8 * 128);
  aliasfree_upfirdn_fused<<<grid, dim3(256), 0, stream>>>(x, bias, fu, fd, out);

  (void)in_sizes; (void)n_in; (void)out_size; (void)d_ws; (void)ws_size;
}